// attn_79920751444354
// MI455X (gfx1250) — compile-verified
//
#include <hip/hip_runtime.h>
#include <hip/hip_bf16.h>
#include <math.h>

#define BB 64
#define TT 512
#define HH 1024

typedef __attribute__((ext_vector_type(2))) float v2f;
typedef __attribute__((ext_vector_type(8))) float v8f;

// K1: h = h0 + h1 ; c[b] = b_lin . h[b]
__global__ void k_prep(const float* __restrict__ h0, const float* __restrict__ h1,
                       const float* __restrict__ b_lin,
                       float* __restrict__ ws_h, float* __restrict__ ws_c) {
    int b = blockIdx.x, tid = threadIdx.x;
    __shared__ float red[256];
    float p = 0.f;
    for (int i = tid; i < HH; i += 256) {
        float v = h0[b * HH + i] + h1[b * HH + i];
        ws_h[b * HH + i] = v;
        p += b_lin[i] * v;
    }
    red[tid] = p; __syncthreads();
    for (int s = 128; s > 0; s >>= 1) {
        if (tid < s) red[tid] += red[tid + s];
        __syncthreads();
    }
    if (tid == 0) ws_c[b] = red[0];
}

// K2: v = h @ W_lin  (M=64, K=1024, N=1024), fp32 WMMA 16x16x4.
// One wave computes one 16x16 tile. A layout (ISA 7.12.2, 32-bit A 16x4):
// lanes 0-15: M=lane, VGPR0=K0,VGPR1=K1 ; lanes 16-31: M=lane-16, VGPR0=K2,VGPR1=K3.
__global__ void k_vgemm(const float* __restrict__ ws_h, const float* __restrict__ W_lin,
                        float* __restrict__ ws_v) {
    int lane = threadIdx.x;
    int l16 = lane & 15;
    int half = lane >> 4;
    int col = blockIdx.x * 16 + l16;   // N index (output h')
    int row = blockIdx.y * 16 + l16;   // M index (batch)
    v8f acc = {};
    const float* arow = ws_h + (size_t)row * HH;
    for (int k0 = 0; k0 < HH; k0 += 4) {
        int k = k0 + 2 * half;
        v2f a = *(const v2f*)(arow + k);              // consecutive K pair, 8B aligned
        v2f bm;
        bm.x = W_lin[(size_t)k * HH + col];           // B[k][n]   = W_lin[k, n]
        bm.y = W_lin[(size_t)(k + 1) * HH + col];     // B[k+1][n]
        acc = __builtin_amdgcn_wmma_f32_16x16x4_f32(
            false, a, false, bm, (short)0, acc, false, false);
    }
    // D layout: lanes 0-15 -> M=i, N=lane ; lanes 16-31 -> M=8+i, N=lane-16
    int outRow = blockIdx.y * 16 + half * 8;
#pragma unroll
    for (int i = 0; i < 8; ++i)
        ws_v[(size_t)(outRow + i) * HH + col] = acc[i];
}

// K3: logits[b,t] = tanh(enc[b,t,:] . v[b,:] + c[b]).  Pass 1 over enc.
__global__ void k_logits(const float* __restrict__ enc, const float* __restrict__ ws_v,
                         const float* __restrict__ ws_c, float* __restrict__ ws_logits) {
    int b = blockIdx.y;
    __shared__ float v_sh[HH];
    for (int i = threadIdx.x; i < HH; i += 256) v_sh[i] = ws_v[(size_t)b * HH + i];
    __syncthreads();
    float cb = ws_c[b];
    int wave = threadIdx.x >> 5, lane = threadIdx.x & 31;
#pragma unroll 1
    for (int r = 0; r < 8; ++r) {
        int t = blockIdx.x * 64 + wave * 8 + r;
        const float* erow = enc + ((size_t)b * TT + t) * HH;
        float acc = 0.f;
        for (int j = lane; j < HH; j += 32) acc += erow[j] * v_sh[j];  // 128B/wave-iter
        for (int off = 16; off > 0; off >>= 1) acc += __shfl_xor(acc, off, 32);
        if (lane == 0) ws_logits[(size_t)b * TT + t] = tanhf(acc + cb);
    }
}

// K4: softmax over T per batch; writes scores directly into d_out slot.
__global__ void k_softmax(const float* __restrict__ ws_logits, float* __restrict__ scores) {
    int b = blockIdx.x, tid = threadIdx.x;
    __shared__ float red[256];
    float l0 = ws_logits[(size_t)b * TT + tid];
    float l1 = ws_logits[(size_t)b * TT + tid + 256];
    red[tid] = fmaxf(l0, l1); __syncthreads();
    for (int s = 128; s > 0; s >>= 1) {
        if (tid < s) red[tid] = fmaxf(red[tid], red[tid + s]);
        __syncthreads();
    }
    float M = red[0]; __syncthreads();
    float e0 = expf(l0 - M), e1 = expf(l1 - M);
    red[tid] = e0 + e1; __syncthreads();
    for (int s = 128; s > 0; s >>= 1) {
        if (tid < s) red[tid] += red[tid + s];
        __syncthreads();
    }
    float inv = 1.f / red[0];
    scores[(size_t)b * TT + tid]       = e0 * inv;
    scores[(size_t)b * TT + tid + 256] = e1 * inv;
}

// K5: out_sum[b,h] = sum_t s*enc ; loss partials sum (enc*(1-s))^2.  Pass 2 over enc.
__global__ void k_outsum(const float* __restrict__ enc, const float* __restrict__ scores,
                         float* __restrict__ ws_outsum, float* __restrict__ ws_lossp) {
    int b = blockIdx.y;
    int h = blockIdx.x * 256 + threadIdx.x;
    __shared__ float s_sh[TT];
    __shared__ float red[256];
    for (int t = threadIdx.x; t < TT; t += 256) s_sh[t] = scores[(size_t)b * TT + t];
    __syncthreads();
    float acc = 0.f, lacc = 0.f;
    const float* eb = enc + (size_t)b * TT * HH + h;
    for (int t = 0; t < TT; ++t) {
        float e = eb[(size_t)t * HH];   // consecutive threads -> consecutive h: coalesced
        float s = s_sh[t];
        acc += s * e;
        float d = e * (1.f - s);
        lacc += d * d;
    }
    ws_outsum[(size_t)b * HH + h] = acc;
    red[threadIdx.x] = lacc; __syncthreads();
    for (int s2 = 128; s2 > 0; s2 >>= 1) {
        if (threadIdx.x < s2) red[threadIdx.x] += red[threadIdx.x + s2];
        __syncthreads();
    }
    if (threadIdx.x == 0) ws_lossp[blockIdx.y * 4 + blockIdx.x] = red[0];
}

// K6: out = tanh([out_sum | h] @ W_out^T + b_out) via fp32 WMMA (K=2048),
//     plus deterministic loss finalization from 256 per-block partials.
__global__ void k_outgemm(const float* __restrict__ ws_outsum, const float* __restrict__ ws_h,
                          const float* __restrict__ W_out, const float* __restrict__ b_out,
                          const float* __restrict__ ws_lossp,
                          float* __restrict__ out, float* __restrict__ loss) {
    int lane = threadIdx.x;
    int l16 = lane & 15, half = lane >> 4;
    int col = blockIdx.x * 16 + l16;   // output feature o
    int row = blockIdx.y * 16 + l16;   // batch b
    if (blockIdx.x == 0 && blockIdx.y == 0 && lane == 0) {
        float s = 0.f;
        for (int i = 0; i < 256; ++i) s += ws_lossp[i];   // fixed order -> deterministic
        loss[0] = s * (1.f / ((float)BB * (float)TT * (float)HH));
    }
    v8f acc = {};
    const float* arow0 = ws_outsum + (size_t)row * HH;
    const float* arow1 = ws_h + (size_t)row * HH;
    const float* brow  = W_out + (size_t)col * (2 * HH);  // B[k][n] = W_out[n, k]
    for (int k0 = 0; k0 < 2 * HH; k0 += 4) {
        int k = k0 + 2 * half;   // boundary at 1024 never straddles (k0 % 4 == 0)
        const float* ap = (k < HH) ? (arow0 + k) : (arow1 + (k - HH));
        v2f a  = *(const v2f*)ap;
        v2f bm = *(const v2f*)(brow + k);                 // consecutive K pair
        acc = __builtin_amdgcn_wmma_f32_16x16x4_f32(
            false, a, false, bm, (short)0, acc, false, false);
    }
    int outRow = blockIdx.y * 16 + half * 8;
    float bias = b_out[col];
#pragma unroll
    for (int i = 0; i < 8; ++i)
        out[(size_t)(outRow + i) * HH + col] = tanhf(acc[i] + bias);
}

extern "C" void kernel_launch(void* const* d_in, const int* in_sizes, int n_in,
                              void* d_out, int out_size, void* d_ws, size_t ws_size,
                              hipStream_t stream) {
    const float* hidden0 = (const float*)d_in[0];
    const float* hidden1 = (const float*)d_in[1];
    const float* enc     = (const float*)d_in[2];
    const float* W_lin   = (const float*)d_in[3];
    const float* b_lin   = (const float*)d_in[4];
    const float* W_out   = (const float*)d_in[5];
    const float* b_out   = (const float*)d_in[6];

    float* out    = (float*)d_out;        // [B,H]   = 65536
    float* loss   = out + BB * HH;        // scalar
    float* scores = loss + 1;             // [B,T,1] = 32768

    float* ws         = (float*)d_ws;
    float* ws_h       = ws;                       // B*H
    float* ws_c       = ws_h + BB * HH;           // B
    float* ws_v       = ws_c + BB;                // B*H
    float* ws_logits  = ws_v + BB * HH;           // B*T
    float* ws_outsum  = ws_logits + BB * TT;      // B*H
    float* ws_lossp   = ws_outsum + BB * HH;      // 256
    (void)in_sizes; (void)n_in; (void)out_size; (void)ws_size;

    k_prep   <<<BB, 256, 0, stream>>>(hidden0, hidden1, b_lin, ws_h, ws_c);
    k_vgemm  <<<dim3(HH / 16, BB / 16), 32, 0, stream>>>(ws_h, W_lin, ws_v);
    k_logits <<<dim3(TT / 64, BB), 256, 0, stream>>>(enc, ws_v, ws_c, ws_logits);
    k_softmax<<<BB, 256, 0, stream>>>(ws_logits, scores);
    k_outsum <<<dim3(HH / 256, BB), 256, 0, stream>>>(enc, scores, ws_outsum, ws_lossp);
    k_outgemm<<<dim3(HH / 16, BB / 16), 32, 0, stream>>>(ws_outsum, ws_h, W_out, b_out,
                                                         ws_lossp, out, loss);
}